// HANLayer_58162447123176
// MI455X (gfx1250) — compile-verified
//
#include <hip/hip_runtime.h>
#include <hip/hip_bf16.h>

// ---------------------------------------------------------------------------
// HAN layer for MI455X (gfx1250, wave32, WMMA).
// Dense GEMMs via V_WMMA_F32_16X16X4_F32 (fp32-exact; GEMMs ~28 GFLOP total,
// negligible vs ~7.5 GB of edge gather/scatter traffic at 23.3 TB/s).
// Edge softmax/aggregation: coalesced per-(edge,feature) threads, float4
// gathers, f32 atomics (stream through 192 MB L2 - all node tensors fit).
// ---------------------------------------------------------------------------

#define NN   50000
#define EE   800000
#define MM   3
#define INF_ 256      // input feature size
#define WSZ  128      // w_h output size
#define HID_ 128      // layer-0 hidden (1 head x 128)
#define HH   8        // layer-1 heads
#define DD   32       // layer-1 dim per head
#define HD   (HH*DD)  // 256
#define SEM  128
#define SLOPE 0.2f

typedef float v8f __attribute__((ext_vector_type(8)));
typedef float v2f __attribute__((ext_vector_type(2)));

// ---------------------------------------------------------------------------
// fp32 WMMA GEMM: C[Mr x Nc] = A[Mr x K] * B[K x Nc], row-major.
// One wave computes a 16x64 strip (4 x 16x16 tiles, A fragment reused 4x).
// REQUIREMENTS: Mr % 16 == 0 (50000 = 3125*16 here), K % 4 == 0, Nc % 64 == 0.
// No bounds checks -> branch-free inner loop (loads + 4 WMMA per k-step).
// A 16x4 fragment: lanes 0-15 carry K={0,1}, lanes 16-31 carry K={2,3}
//   -> one 8-byte vector load per lane per k-step.
// B 4x16 fragment: VGPR0 = rows K0(l0-15)/K2(l16-31), VGPR1 = K1/K3.
// ---------------------------------------------------------------------------
__global__ __launch_bounds__(256) void gemm16x64_wmma_f32(
    const float* __restrict__ A, const float* __restrict__ B,
    float* __restrict__ C, int Mr, int K, int Nc) {
  int wave   = blockIdx.x * 8 + (int)(threadIdx.x >> 5);
  int strips = Nc >> 6;
  int Mtiles = Mr >> 4;
  int tileM  = wave / strips;
  if (tileM >= Mtiles) return;                  // wave-uniform early out
  int strip  = wave - tileM * strips;

  int lane = threadIdx.x & 31;
  int half = lane >> 4;                          // 0: K0/K1, 1: K2/K3
  int l16  = lane & 15;

  int row = tileM * 16 + l16;
  const float* Arow = A + (size_t)row * K + half * 2;          // 8B aligned
  const float* Bp   = B + (size_t)(half * 2) * Nc + strip * 64 + l16;

  v8f acc0 = {}; v8f acc1 = {}; v8f acc2 = {}; v8f acc3 = {};

  for (int kk = 0; kk < K; kk += 4) {
    v2f a = *(const v2f*)(Arow + kk);            // A[row][kk+2h .. kk+2h+1]

    const float* B0 = Bp;                        // row kk+2h
    const float* B1 = Bp + Nc;                   // row kk+2h+1
    v2f b0; b0.x = B0[0];  b0.y = B1[0];
    v2f b1; b1.x = B0[16]; b1.y = B1[16];
    v2f b2; b2.x = B0[32]; b2.y = B1[32];
    v2f b3; b3.x = B0[48]; b3.y = B1[48];
    Bp += 4 * (size_t)Nc;

    acc0 = __builtin_amdgcn_wmma_f32_16x16x4_f32(false, a, false, b0, (short)0, acc0, false, false);
    acc1 = __builtin_amdgcn_wmma_f32_16x16x4_f32(false, a, false, b1, (short)0, acc1, false, false);
    acc2 = __builtin_amdgcn_wmma_f32_16x16x4_f32(false, a, false, b2, (short)0, acc2, false, false);
    acc3 = __builtin_amdgcn_wmma_f32_16x16x4_f32(false, a, false, b3, (short)0, acc3, false, false);
  }

  // D layout: VGPR v -> row tileM*16 + v + half*8, col = strip*64 + l16 (+16k)
  for (int v = 0; v < 8; ++v) {
    int r = tileM * 16 + v + half * 8;
    float* Crow = C + (size_t)r * Nc + strip * 64 + l16;
    Crow[0]  = acc0[v];
    Crow[16] = acc1[v];
    Crow[32] = acc2[v];
    Crow[48] = acc3[v];
  }
}

// ---------------------------------------------------------------------------
__global__ void fill_u32(unsigned* __restrict__ p, unsigned v, int n) {
  int t = blockIdx.x * blockDim.x + threadIdx.x;
  if (t < n) p[t] = v;
}

// el[n,h] = sum_d feat[n,h,d]*al[h,d]; er likewise.
__global__ void compute_elr(const float* __restrict__ feat,
                            const float* __restrict__ al,
                            const float* __restrict__ ar,
                            float* __restrict__ el, float* __restrict__ er,
                            int Nn, int Hh, int Dd) {
  int t = blockIdx.x * blockDim.x + threadIdx.x;
  if (t >= Nn * Hh) return;
  int n = t / Hh, h = t - n * Hh;
  const float* f = feat + (size_t)n * Hh * Dd + h * Dd;
  const float* a = al + h * Dd;
  const float* r = ar + h * Dd;
  float sl = 0.f, sr = 0.f;
  for (int d = 0; d < Dd; ++d) { float x = f[d]; sl += x * a[d]; sr += x * r[d]; }
  el[t] = sl; er[t] = sr;
}

__device__ __forceinline__ float lrelu(float x) { return x > 0.f ? x : SLOPE * x; }

// Monotone float atomic-max via signed/unsigned int atomics (init 0xff800000).
__device__ __forceinline__ void atomicMaxF32(float* addr, float v) {
  if (v >= 0.f) atomicMax((int*)addr, __float_as_int(v));
  else          atomicMin((unsigned*)addr, (unsigned)__float_as_int(v));
}

__global__ void edge_logits_max(const int* __restrict__ src, const int* __restrict__ dst,
                                const float* __restrict__ el, const float* __restrict__ er,
                                float* __restrict__ emax, int Ee, int Hh) {
  int t = blockIdx.x * blockDim.x + threadIdx.x;
  if (t >= Ee * Hh) return;
  int e = t / Hh, h = t - e * Hh;
  float x = lrelu(el[src[e] * Hh + h] + er[dst[e] * Hh + h]);
  atomicMaxF32(&emax[dst[e] * Hh + h], x);
}

__global__ void edge_exp_sum(const int* __restrict__ src, const int* __restrict__ dst,
                             const float* __restrict__ el, const float* __restrict__ er,
                             const float* __restrict__ emax,
                             float* __restrict__ aE, float* __restrict__ denom,
                             int Ee, int Hh) {
  int t = blockIdx.x * blockDim.x + threadIdx.x;
  if (t >= Ee * Hh) return;
  int e = t / Hh, h = t - e * Hh;
  int dh = dst[e] * Hh + h;
  float x = lrelu(el[src[e] * Hh + h] + er[dh]);
  float a = __expf(x - emax[dh]);
  aE[t] = a;
  atomicAdd(&denom[dh], a);
}

// agg[dst, f] += a[e, h(f)] * feat[src, f], 4 floats per thread:
// float4 gather (global_load_b128) + 4 f32 atomic adds. A wave shares
// (src,dst) -> broadcast index load, fully coalesced rows.
__global__ void edge_aggregate4(const int* __restrict__ src, const int* __restrict__ dst,
                                const float* __restrict__ aE, const float4* __restrict__ feat4,
                                float* __restrict__ agg, int Ee, int Hd4, int Dd4) {
  int t = blockIdx.x * blockDim.x + threadIdx.x;
  if (t >= Ee * Hd4) return;
  int e = t / Hd4, q = t - e * Hd4;          // q: float4 index within row
  int h = q / Dd4;
  float a = aE[e * (Hd4 / Dd4) + h];
  float4 f = feat4[(size_t)src[e] * Hd4 + q];
  float* o = agg + ((size_t)dst[e] * Hd4 + q) * 4;
  atomicAdd(o + 0, a * f.x);
  atomicAdd(o + 1, a * f.y);
  atomicAdd(o + 2, a * f.z);
  atomicAdd(o + 3, a * f.w);
}

// out = elu(agg / max(denom,!=0) + bias) in place.
__global__ void finalize_elu(float* __restrict__ agg, const float* __restrict__ denom,
                             const float* __restrict__ bias, int Nn, int Hd, int Dd) {
  int t = blockIdx.x * blockDim.x + threadIdx.x;
  if (t >= Nn * Hd) return;
  int n = t / Hd, f = t - n * Hd;
  float dn = denom[n * (Hd / Dd) + f / Dd];
  dn = (dn == 0.f) ? 1.f : dn;
  float v = agg[t] / dn + bias[f];
  agg[t] = v > 0.f ? v : expm1f(v);
}

// w[n,m] = sum_j tanh(t[n,j] + bs1[j]) * Ws2[j]
__global__ void semantic_score(const float* __restrict__ t, const float* __restrict__ bs1,
                               const float* __restrict__ Ws2, float* __restrict__ w,
                               int Nn, int m) {
  int n = blockIdx.x * blockDim.x + threadIdx.x;
  if (n >= Nn) return;
  const float* row = t + (size_t)n * SEM;
  float acc = 0.f;
  for (int j = 0; j < SEM; ++j) acc += tanhf(row[j] + bs1[j]) * Ws2[j];
  w[n * MM + m] = acc;
}

// out[n,c] = sum_m softmax_m(w[n,:]) * z[m,n,c]
__global__ void semantic_combine(const float* __restrict__ z, const float* __restrict__ w,
                                 float* __restrict__ out, int Nn) {
  int t = blockIdx.x * blockDim.x + threadIdx.x;
  if (t >= Nn * HD) return;
  int n = t / HD, c = t - n * HD;
  float w0 = w[n * MM + 0], w1 = w[n * MM + 1], w2 = w[n * MM + 2];
  float mx = fmaxf(w0, fmaxf(w1, w2));
  float e0 = __expf(w0 - mx), e1 = __expf(w1 - mx), e2 = __expf(w2 - mx);
  float inv = 1.f / (e0 + e1 + e2);
  size_t idx = (size_t)n * HD + c;
  const size_t ZS = (size_t)NN * HD;
  out[idx] = (e0 * z[idx] + e1 * z[ZS + idx] + e2 * z[2 * ZS + idx]) * inv;
}

// ---------------------------------------------------------------------------
extern "C" void kernel_launch(void* const* d_in, const int* in_sizes, int n_in,
                              void* d_out, int out_size, void* d_ws, size_t ws_size,
                              hipStream_t stream) {
  const float* h     = (const float*)d_in[0];
  const int*   edges = (const int*)  d_in[1];
  const float* Wh    = (const float*)d_in[2];
  const float* W0    = (const float*)d_in[3];
  const float* al0   = (const float*)d_in[4];
  const float* ar0   = (const float*)d_in[5];
  const float* b0    = (const float*)d_in[6];
  const float* W1    = (const float*)d_in[7];
  const float* al1   = (const float*)d_in[8];
  const float* ar1   = (const float*)d_in[9];
  const float* b1    = (const float*)d_in[10];
  const float* Ws1   = (const float*)d_in[11];
  const float* bs1   = (const float*)d_in[12];
  const float* Ws2   = (const float*)d_in[13];
  float* out = (float*)d_out;

  // Workspace layout (~289 MB of fp32):
  float* ws    = (float*)d_ws;
  float* wh    = ws;                 ws += (size_t)NN * WSZ;      // [N,128]
  float* z     = ws;                 ws += (size_t)MM * NN * HD;  // [3,N,256]
  float* featX = ws;                 ws += (size_t)NN * HD;       // [N,256] scratch
  float* y0    = ws;                 ws += (size_t)NN * HID_;     // [N,128]
  float* el    = ws;                 ws += (size_t)NN * HH;
  float* er    = ws;                 ws += (size_t)NN * HH;
  float* emax  = ws;                 ws += (size_t)NN * HH;
  float* denom = ws;                 ws += (size_t)NN * HH;
  float* aE    = ws;                 ws += (size_t)EE * HH;
  float* wsem  = ws;                 ws += (size_t)NN * MM;

  const int B = 256;
  auto blocks = [](long long n, int b) { return (unsigned)((n + b - 1) / b); };
  auto gemm = [&](const float* A, const float* Bm, float* C, int Mr, int K, int Nc) {
    int waves = (Mr / 16) * (Nc / 64);
    gemm16x64_wmma_f32<<<dim3((waves + 7) / 8), dim3(256), 0, stream>>>(A, Bm, C, Mr, K, Nc);
  };

  // wh = h @ Wh
  gemm(h, Wh, wh, NN, INF_, WSZ);

  for (int m = 0; m < MM; ++m) {
    const int* src = edges + (size_t)m * 2 * EE;
    const int* dst = src + EE;

    // ---- GAT layer 0: wh -> y0 (1 head x 128) ----
    gemm(wh, W0 + (size_t)m * WSZ * HID_, featX, NN, WSZ, HID_);
    compute_elr<<<blocks(NN * 1, B), B, 0, stream>>>(featX, al0 + m * HID_, ar0 + m * HID_,
                                                     el, er, NN, 1, HID_);
    fill_u32<<<blocks(NN, B), B, 0, stream>>>((unsigned*)emax, 0xff800000u, NN);       // -inf
    fill_u32<<<blocks(NN, B), B, 0, stream>>>((unsigned*)denom, 0u, NN);
    fill_u32<<<blocks((long long)NN * HID_, B), B, 0, stream>>>((unsigned*)y0, 0u, NN * HID_);
    edge_logits_max<<<blocks(EE, B), B, 0, stream>>>(src, dst, el, er, emax, EE, 1);
    edge_exp_sum<<<blocks(EE, B), B, 0, stream>>>(src, dst, el, er, emax, aE, denom, EE, 1);
    edge_aggregate4<<<blocks((long long)EE * (HID_ / 4), B), B, 0, stream>>>(
        src, dst, aE, (const float4*)featX, y0, EE, HID_ / 4, HID_ / 4);
    finalize_elu<<<blocks((long long)NN * HID_, B), B, 0, stream>>>(y0, denom, b0 + m * HID_,
                                                                    NN, HID_, HID_);

    // ---- GAT layer 1: y0 -> z[m] (8 heads x 32) ----
    float* zm = z + (size_t)m * NN * HD;
    gemm(y0, W1 + (size_t)m * HID_ * HD, featX, NN, HID_, HD);
    compute_elr<<<blocks(NN * HH, B), B, 0, stream>>>(featX, al1 + m * HD, ar1 + m * HD,
                                                      el, er, NN, HH, DD);
    fill_u32<<<blocks(NN * HH, B), B, 0, stream>>>((unsigned*)emax, 0xff800000u, NN * HH);
    fill_u32<<<blocks(NN * HH, B), B, 0, stream>>>((unsigned*)denom, 0u, NN * HH);
    fill_u32<<<blocks((long long)NN * HD, B), B, 0, stream>>>((unsigned*)zm, 0u, NN * HD);
    edge_logits_max<<<blocks((long long)EE * HH, B), B, 0, stream>>>(src, dst, el, er, emax, EE, HH);
    edge_exp_sum<<<blocks((long long)EE * HH, B), B, 0, stream>>>(src, dst, el, er, emax, aE, denom,
                                                                  EE, HH);
    edge_aggregate4<<<blocks((long long)EE * (HD / 4), B), B, 0, stream>>>(
        src, dst, aE, (const float4*)featX, zm, EE, HD / 4, DD / 4);
    finalize_elu<<<blocks((long long)NN * HD, B), B, 0, stream>>>(zm, denom, b1 + m * HD,
                                                                  NN, HD, DD);
  }

  // ---- semantic attention ----
  for (int m = 0; m < MM; ++m) {
    gemm(z + (size_t)m * NN * HD, Ws1, featX, NN, HD, SEM);   // t = z_m @ Ws1
    semantic_score<<<blocks(NN, B), B, 0, stream>>>(featX, bs1, Ws2, wsem, NN, m);
  }
  semantic_combine<<<blocks((long long)NN * HD, B), B, 0, stream>>>(z, wsem, out, NN);
}